// TopKSAE_81767587381716
// MI455X (gfx1250) — compile-verified
//
#include <hip/hip_runtime.h>
#include <hip/hip_bf16.h>

// ---------------------------------------------------------------------------
// TopK SAE forward for MI455X (gfx1250), fp32 WMMA path.
//   B=4, T=1024 -> BT=4096 tokens;  E=768, F=24576, K=32
//   encode: latent = (x - b_dec) @ W_enc + b_enc        (fp32 WMMA 16x16x4)
//   topk  : keep top-32 per token (running register top-k, no latent HBM R/T)
//   decode: x_rec = b_dec + sum_i v_i * W_dec[f_i,:]    (sparse, L2-cached)
// d_out = [x_rec (4096*768) | feature_magnitudes (4096*24576)]
// ---------------------------------------------------------------------------

typedef __attribute__((ext_vector_type(2))) float v2f;
typedef __attribute__((ext_vector_type(8))) float v8f;

#define BT      4096
#define EDIM    768
#define FDIM    24576
#define KSEL    32

#define M_TILE  128      // tokens per block
#define N_SLAB  1024     // features per block (column slab)
#define N_SUB   64       // features per inner sub-tile
#define KC      32       // K-chunk staged in LDS
#define NCHUNKS (FDIM / N_SLAB)   // 24
#define NSUBS   (N_SLAB / N_SUB)  // 16
#define KCHUNKS (EDIM / KC)       // 24

#define A_STRIDE 36      // 32 + 4 pad: rows 144 B -> b128-aligned LDS stores
#define L_STRIDE 66      // 64 + 2 pad

__device__ __forceinline__ float wave_min32(float v) {
#pragma unroll
  for (int off = 16; off > 0; off >>= 1) v = fminf(v, __shfl_xor(v, off));
  return v;
}
__device__ __forceinline__ float wave_max32(float v) {
#pragma unroll
  for (int off = 16; off > 0; off >>= 1) v = fmaxf(v, __shfl_xor(v, off));
  return v;
}

// ===========================================================================
// Kernel 1: tiled fp32 WMMA encode + per-token running top-32 per slab.
// grid = (BT/M_TILE=32, NCHUNKS=24), block = 256 (8 waves; wave w owns the
// 16-token stripe [w*16, w*16+16) for both GEMM rows and top-k lists).
// ===========================================================================
__global__ __launch_bounds__(256) void sae_encode_topk(
    const float* __restrict__ x, const float* __restrict__ W_enc,
    const float* __restrict__ b_enc, const float* __restrict__ b_dec,
    float* __restrict__ wsVal, int* __restrict__ wsIdx) {
  __shared__ float As[M_TILE * A_STRIDE];       // 128x32 A chunk (x - b_dec)
  __shared__ float Bt[N_SUB * A_STRIDE];        // 64x32 B chunk, transposed
  __shared__ float Lat[8 * 16 * L_STRIDE];      // per-wave 16x64 latent tile

  const int tid  = threadIdx.x;
  const int lane = tid & 31;
  const int wid  = tid >> 5;
  const int kh   = lane >> 4;    // K half-select (WMMA fp32 layout)
  const int l16  = lane & 15;

  const int tokBase  = blockIdx.x * M_TILE;
  const int slabBase = blockIdx.y * N_SLAB;

  // Staging coordinates (i-invariant: 256 % 8 == 256 % 16 == 0)
  const int aM0 = tid >> 3;           // A row for i=0 (+32 per i)
  const int aK4 = (tid & 7) * 4;      // A k-quad (constant per thread)
  const int bK0 = tid >> 4;           // B k for i=0 (+16 per i)
  const int bN4 = (tid & 15) * 4;     // B n-quad (constant per thread)

  // running top-32 per owned token: one slot per lane
  float candV[16];
  int   candI[16];
  float runMin[16];
#pragma unroll
  for (int t = 0; t < 16; ++t) {
    candV[t] = -__builtin_inff();
    candI[t] = 0;
    runMin[t] = -__builtin_inff();
  }

  float* latW = &Lat[wid * 16 * L_STRIDE];
  const int aRow = (wid * 16 + l16) * A_STRIDE;

  for (int ns = 0; ns < NSUBS; ++ns) {
    const int nBase = slabBase + ns * N_SUB;

    v8f acc[4];
#pragma unroll
    for (int nf = 0; nf < 4; ++nf)
#pragma unroll
      for (int r = 0; r < 8; ++r) acc[nf][r] = 0.0f;

    for (int kc = 0; kc < KCHUNKS; ++kc) {
      const int kBase = kc * KC;
      __syncthreads();

      // ---- load phase: issue all global b128 loads back-to-back ----
      float4 bd = *(const float4*)&b_dec[kBase + aK4];
      float4 aR[4];
#pragma unroll
      for (int i = 0; i < 4; ++i)
        aR[i] = *(const float4*)
            &x[(size_t)(tokBase + aM0 + i * 32) * EDIM + kBase + aK4];
      float4 bR[2];
#pragma unroll
      for (int i = 0; i < 2; ++i)
        bR[i] = *(const float4*)
            &W_enc[(size_t)(kBase + bK0 + i * 16) * FDIM + nBase + bN4];

      // prefetch next K-chunk (global_prefetch_b8) behind this chunk's math
      if (kc + 1 < KCHUNKS) {
        __builtin_prefetch(
            &W_enc[(size_t)(kBase + KC + bK0) * FDIM + nBase + bN4], 0, 1);
        __builtin_prefetch(
            &x[(size_t)(tokBase + aM0) * EDIM + kBase + KC + aK4], 0, 1);
      }

      // ---- store phase: A as b128 rows, B transposed as b32 ----
#pragma unroll
      for (int i = 0; i < 4; ++i) {
        float4 v = aR[i];
        *(float4*)&As[(aM0 + i * 32) * A_STRIDE + aK4] =
            make_float4(v.x - bd.x, v.y - bd.y, v.z - bd.z, v.w - bd.w);
      }
#pragma unroll
      for (int i = 0; i < 2; ++i) {
        const int k = bK0 + i * 16;
        Bt[(bN4 + 0) * A_STRIDE + k] = bR[i].x;
        Bt[(bN4 + 1) * A_STRIDE + k] = bR[i].y;
        Bt[(bN4 + 2) * A_STRIDE + k] = bR[i].z;
        Bt[(bN4 + 3) * A_STRIDE + k] = bR[i].w;
      }
      __syncthreads();

#pragma unroll
      for (int kk = 0; kk < 8; ++kk) {
        const int ko = kk * 4 + kh * 2;
        // A 16x4 frag: lane (m=l16, khalf=kh) holds K-pair {ko, ko+1}
        v2f a = *(const v2f*)&As[aRow + ko];
#pragma unroll
        for (int nf = 0; nf < 4; ++nf) {
          // B 4x16 frag: lane (n=l16, khalf=kh) holds K-pair {ko, ko+1}
          v2f b = *(const v2f*)&Bt[(nf * 16 + l16) * A_STRIDE + ko];
          acc[nf] = __builtin_amdgcn_wmma_f32_16x16x4_f32(
              false, a, false, b, (short)0, acc[nf], false, false);
        }
      }
    }

    // latent tile (+ b_enc) -> per-wave LDS stripe
#pragma unroll
    for (int nf = 0; nf < 4; ++nf) {
      float be = b_enc[nBase + nf * 16 + l16];
#pragma unroll
      for (int r = 0; r < 8; ++r) {
        latW[(r + 8 * kh) * L_STRIDE + nf * 16 + l16] = acc[nf][r] + be;
      }
    }
    __syncthreads();

    // running top-32 update: ballot candidates > current kth, serial insert
#pragma unroll
    for (int t = 0; t < 16; ++t) {
#pragma unroll
      for (int j = 0; j < 2; ++j) {
        float v = latW[t * L_STRIDE + lane + j * 32];
        unsigned mask = (unsigned)__ballot(v > runMin[t]);
        while (mask) {
          int L = __ffs(mask) - 1;
          mask &= mask - 1;
          float vL = __shfl(v, L);
          if (vL > runMin[t]) {
            float mn = wave_min32(candV[t]);
            unsigned am = (unsigned)__ballot(candV[t] == mn);
            int aml = __ffs(am) - 1;
            if (lane == aml) {
              candV[t] = vL;
              candI[t] = nBase + L + j * 32;
            }
            runMin[t] = wave_min32(candV[t]);
          }
        }
      }
    }
    __syncthreads();
  }

  // emit this slab's 32 candidates per owned token
#pragma unroll
  for (int t = 0; t < 16; ++t) {
    int token = tokBase + wid * 16 + t;
    size_t base = ((size_t)token * NCHUNKS + blockIdx.y) * KSEL + lane;
    wsVal[base] = candV[t];
    wsIdx[base] = candI[t];
  }
}

// ===========================================================================
// Kernel 2: merge 24x32 candidates -> true top-32, write fm (zeros+scatter),
// sparse decode x_rec = b_dec + sum v_i * W_dec[f_i,:].  One block per token.
// ===========================================================================
__global__ __launch_bounds__(256) void sae_select_decode(
    const float* __restrict__ wsVal, const int* __restrict__ wsIdx,
    const float* __restrict__ W_dec, const float* __restrict__ b_dec,
    float* __restrict__ xrec, float* __restrict__ fm) {
  __shared__ float cV[NCHUNKS * KSEL];   // 768
  __shared__ int   cI[NCHUNKS * KSEL];
  __shared__ float topV[KSEL];
  __shared__ int   topI[KSEL];

  const int tid  = threadIdx.x;
  const int lane = tid & 31;
  const int wid  = tid >> 5;
  const int token = blockIdx.x;

  for (int i = tid; i < NCHUNKS * KSEL; i += 256) {
    cV[i] = wsVal[(size_t)token * (NCHUNKS * KSEL) + i];
    cI[i] = wsIdx[(size_t)token * (NCHUNKS * KSEL) + i];
  }
  __syncthreads();

  if (wid == 0) {
    for (int r = 0; r < KSEL; ++r) {
      float lm = -__builtin_inff();
      int la = -1;
#pragma unroll
      for (int i = 0; i < NCHUNKS; ++i) {
        float v = cV[lane + i * 32];
        if (v > lm) { lm = v; la = lane + i * 32; }
      }
      float wm = wave_max32(lm);
      unsigned msk = (unsigned)__ballot(lm == wm);
      int win = __ffs(msk) - 1;
      if (lane == win) {
        topV[r] = wm;
        topI[r] = cI[la];
        cV[la] = -__builtin_inff();   // remove for next round
      }
    }
  }
  __syncthreads();

  // fm row: zero-fill then scatter survivors
  float4* fmrow = (float4*)(fm + (size_t)token * FDIM);
#pragma unroll
  for (int i = 0; i < FDIM / 4 / 256; ++i)
    fmrow[tid + i * 256] = make_float4(0.f, 0.f, 0.f, 0.f);
  __threadfence();
  __syncthreads();
  if (wid == 0) fm[(size_t)token * FDIM + topI[lane]] = topV[lane];

  // sparse decode (W_dec rows are L2-resident: 75.5 MB < 192 MB L2)
#pragma unroll
  for (int eo = 0; eo < EDIM / 256; ++eo) {
    int e = tid + eo * 256;
    float acc = b_dec[e];
#pragma unroll
    for (int i = 0; i < KSEL; ++i)
      acc += topV[i] * W_dec[(size_t)topI[i] * EDIM + e];
    xrec[(size_t)token * EDIM + e] = acc;
  }
}

// ===========================================================================
extern "C" void kernel_launch(void* const* d_in, const int* in_sizes, int n_in,
                              void* d_out, int out_size, void* d_ws,
                              size_t ws_size, hipStream_t stream) {
  (void)in_sizes; (void)n_in; (void)out_size; (void)ws_size;
  const float* x     = (const float*)d_in[0];
  const float* W_enc = (const float*)d_in[1];
  const float* W_dec = (const float*)d_in[2];
  const float* b_enc = (const float*)d_in[3];
  const float* b_dec = (const float*)d_in[4];
  // d_in[5] = k (==32, compile-time constant here)

  float* xrec = (float*)d_out;
  float* fm   = (float*)d_out + (size_t)BT * EDIM;

  const size_t nCand = (size_t)BT * NCHUNKS * KSEL;  // 3,145,728
  float* wsVal = (float*)d_ws;
  int*   wsIdx = (int*)((char*)d_ws + nCand * sizeof(float));

  dim3 g1(BT / M_TILE, NCHUNKS);   // 32 x 24
  sae_encode_topk<<<g1, 256, 0, stream>>>(x, W_enc, b_enc, b_dec, wsVal, wsIdx);
  sae_select_decode<<<BT, 256, 0, stream>>>(wsVal, wsIdx, W_dec, b_dec, xrec, fm);
}